// AdaptiveMBlock_86543591014676
// MI455X (gfx1250) — compile-verified
//
#include <hip/hip_runtime.h>
#include <hip/hip_bf16.h>

// ---- fixed problem shape from the reference -------------------------------
#define BB 4
#define HH 256
#define WW 256
#define CC 16
#define NPIX (BB * HH * WW)          // 262144
#define PIX_PER_BLOCK 128
#define LDSTRIDE 50                  // 48 features + 2 pad floats (bank spread)

typedef __attribute__((ext_vector_type(2))) float v2f;
typedef __attribute__((ext_vector_type(8))) float v8f;

// compare-exchange (ascending)
__device__ __forceinline__ void cswap(float& a, float& b) {
    float lo = fminf(a, b);
    float hi = fmaxf(a, b);
    a = lo; b = hi;
}

// Partial bubble selection: after (N-IDX) max-bubbling passes, a[IDX] holds the
// ascending-order element at index IDX.  Fully unrolled compare/exchange net.
template <int N, int IDX>
__device__ __forceinline__ float select_kth(float* a) {
#pragma unroll
    for (int p = 0; p < N - IDX; ++p) {
#pragma unroll
        for (int i = 0; i < N - 1 - p; ++i) {
            cswap(a[i], a[i + 1]);
        }
    }
    return a[IDX];
}

// Fused kernel: phase 1 computes the 3 order statistics per (pixel,channel)
// into an LDS feature tile; phase 2 runs the 1x1 conv as WMMA f32 16x16x4.
__global__ void __launch_bounds__(256, 2)
adaptive_mblock_fused(const float* __restrict__ x,
                      const float* __restrict__ w,   // [48][16] row-major
                      float* __restrict__ out) {
    __shared__ float lfeat[PIX_PER_BLOCK * LDSTRIDE];   // 25.6 KB

    const int tid     = threadIdx.x;
    const int pixBase = blockIdx.x * PIX_PER_BLOCK;     // 128 consecutive pixels in a row
    const int b       = pixBase >> 16;                  // / (H*W)
    const int h       = (pixBase >> 8) & (HH - 1);
    const int w0      = pixBase & (WW - 1);             // 0 or 128

    const int c = tid & (CC - 1);

    // ---------------- phase 1: order statistics ----------------
#pragma unroll 1
    for (int it = 0; it < (PIX_PER_BLOCK * CC) / 256; ++it) {
        const int pixLocal = (tid >> 4) + it * 16;      // 0..127
        const int wx = w0 + pixLocal;

        float a49[49];
#pragma unroll
        for (int dy = 0; dy < 7; ++dy) {
#pragma unroll
            for (int dx = 0; dx < 7; ++dx) {
                const int hh = h + dy - 3;
                const int ww = wx + dx - 3;
                float v = 0.0f;
                if (hh >= 0 && hh < HH && ww >= 0 && ww < WW) {
                    v = x[(((b * HH + hh) * WW + ww) * CC) + c];
                }
                a49[dy * 7 + dx] = v;
            }
        }

        // k=3: center 3x3, ascending index 6 of 9
        float a9[9];
#pragma unroll
        for (int dy = 0; dy < 3; ++dy)
#pragma unroll
            for (int dx = 0; dx < 3; ++dx)
                a9[dy * 3 + dx] = a49[(dy + 2) * 7 + (dx + 2)];
        const float m3 = select_kth<9, 6>(a9);

        // k=5: center 5x5, ascending index 14 of 25
        float a25[25];
#pragma unroll
        for (int dy = 0; dy < 5; ++dy)
#pragma unroll
            for (int dx = 0; dx < 5; ++dx)
                a25[dy * 5 + dx] = a49[(dy + 1) * 7 + (dx + 1)];
        const float m5 = select_kth<25, 14>(a25);

        // k=7: full 7x7, ascending index 26 of 49 (destroys a49)
        const float m7 = select_kth<49, 26>(a49);

        // concat order: [m7 | m3 | m5]
        float* row = &lfeat[pixLocal * LDSTRIDE];
        row[c]      = m7;
        row[16 + c] = m3;
        row[32 + c] = m5;
    }

    __syncthreads();

    // ---------------- phase 2: 1x1 conv via WMMA f32 16x16x4 ----------------
    const int lane   = tid & 31;
    const int waveId = tid >> 5;            // 8 waves, 16 pixels each
    const int half   = lane >> 4;           // 0: K-pair {0,1}; 1: K-pair {2,3}
    const int nIdx   = lane & 15;           // N (out channel) / M (pixel row)

    // B matrix: B[K=c][N=o] = w[c*16 + o]
    v2f bmat[12];
#pragma unroll
    for (int kk = 0; kk < 12; ++kk) {
        const int k0 = kk * 4 + half * 2;
        bmat[kk].x = w[(k0 + 0) * 16 + nIdx];
        bmat[kk].y = w[(k0 + 1) * 16 + nIdx];
    }

    // A matrix: A[M=pixel][K=feature] from LDS
    const float* lf = &lfeat[(waveId * 16 + nIdx) * LDSTRIDE + half * 2];

    v8f acc = {0.f, 0.f, 0.f, 0.f, 0.f, 0.f, 0.f, 0.f};
#pragma unroll
    for (int kk = 0; kk < 12; ++kk) {
        v2f a;
        a.x = lf[kk * 4 + 0];
        a.y = lf[kk * 4 + 1];
        // D = A x B + C : v_wmma_f32_16x16x4_f32
        acc = __builtin_amdgcn_wmma_f32_16x16x4_f32(
            /*neg_a=*/false, a, /*neg_b=*/false, bmat[kk],
            /*c_mod=*/(short)0, acc, /*reuse_a=*/false, /*reuse_b=*/false);
    }

    // D layout: VGPR r, lanes 0-15 -> M=r, lanes 16-31 -> M=r+8; N = lane&15
    const int tilePix = pixBase + waveId * 16;
#pragma unroll
    for (int r = 0; r < 8; ++r) {
        const int m = r + half * 8;
        out[(size_t)(tilePix + m) * CC + nIdx] = acc[r];
    }
}

extern "C" void kernel_launch(void* const* d_in, const int* in_sizes, int n_in,
                              void* d_out, int out_size, void* d_ws, size_t ws_size,
                              hipStream_t stream) {
    (void)in_sizes; (void)n_in; (void)d_ws; (void)ws_size; (void)out_size;
    const float* x  = (const float*)d_in[0];                 // [4,256,256,16]
    const float* hk = (const float*)d_in[1];                 // [1,1,48,16] -> [48,16]
    float* out = (float*)d_out;                              // [4,256,256,16]

    const int blocks = NPIX / PIX_PER_BLOCK;                 // 2048
    adaptive_mblock_fused<<<blocks, 256, 0, stream>>>(x, hk, out);
}